// LayerMemN2N_21680994910438
// MI455X (gfx1250) — compile-verified
//
#include <hip/hip_runtime.h>

#define MEM_SIZE 50
#define BATCH    64
#define MAX_LEN  50
#define EMB      256
#define HID      128
#define HOPS     3

typedef __attribute__((ext_vector_type(2))) float v2f;
typedef __attribute__((ext_vector_type(8))) float v8f;

// ---------------------------------------------------------------------------
// Kernel 1: reduce over MAX_LEN (the HBM-bound 164 MB stream).
// rows 0..3199 : insum[jb][e] = sum_l inputs[jb][l][e]
// rows 3200..3263 : qsum[b][e] = sum_l questions[b][l][e]
// 4 rows per 256-thread block; each lane accumulates a float4 (16B) so a
// wave32 issues 512B contiguous loads per iteration.
// ---------------------------------------------------------------------------
__global__ void memn2n_reduce_len(const float* __restrict__ inputs,
                                  const float* __restrict__ questions,
                                  float* __restrict__ insum,
                                  float* __restrict__ qsum)
{
    const int ROWS_IN = MEM_SIZE * BATCH;          // 3200
    int tid = threadIdx.x;                         // 0..255
    int sub = tid >> 6;                            // row-in-block 0..3
    int t   = tid & 63;                            // float4 group 0..63
    int row = blockIdx.x * 4 + sub;                // 0..3263

    const float* src;
    float* dst;
    if (row < ROWS_IN) {
        src = inputs + (size_t)row * (MAX_LEN * EMB);
        dst = insum  + (size_t)row * EMB;
    } else {
        int b = row - ROWS_IN;
        if (b >= BATCH) return;
        src = questions + (size_t)b * (MAX_LEN * EMB);
        dst = qsum      + (size_t)b * EMB;
    }

    const float4* s4 = (const float4*)src;
    float4 acc = make_float4(0.f, 0.f, 0.f, 0.f);
#pragma unroll 5
    for (int l = 0; l < MAX_LEN; ++l) {
        float4 v = s4[l * (EMB / 4) + t];
        acc.x += v.x; acc.y += v.y; acc.z += v.z; acc.w += v.w;
    }
    ((float4*)dst)[t] = acc;
}

// ---------------------------------------------------------------------------
// Kernel 2: f32 WMMA GEMMs (full precision, V_WMMA_F32_16X16X4_F32).
//   m[jb][h] = insum[jb][:] @ A      (3200x256 @ 256x128)
//   c[jb][h] = insum[jb][:] @ C
//   u0[b][h] = qsum[b][:]  @ B      (64x256 @ 256x128)
// One 16x16 output tile per wave; K-loop steps of 4.
// A-frag (16x4 f32): lanes 0-15 -> {K=k,k+1}, lanes 16-31 -> {K=k+2,k+3}, M=lane%16.
// B-frag (4x16 f32): lanes 0-15 -> rows k,k+1, lanes 16-31 -> rows k+2,k+3, N=lane%16.
// C/D (16x16 f32, 8 VGPRs): vg -> M = vg + 8*(lane/16), N = lane%16.
// ---------------------------------------------------------------------------
__global__ void memn2n_wmma_gemm(const float* __restrict__ insum,
                                 const float* __restrict__ qsum,
                                 const float* __restrict__ A,
                                 const float* __restrict__ B,
                                 const float* __restrict__ C,
                                 float* __restrict__ m_ws,
                                 float* __restrict__ c_ws,
                                 float* __restrict__ u_ws)
{
    const int NT  = HID / 16;                        // 8 col tiles
    const int TILES_M = (MEM_SIZE * BATCH / 16) * NT; // 200*8 = 1600
    const int TILES_U = (BATCH / 16) * NT;           // 4*8 = 32

    int wave = (blockIdx.x * blockDim.x + threadIdx.x) >> 5;
    int lane = threadIdx.x & 31;

    const float* Amat;
    const float* W;
    float* out;
    int row0, col0;

    if (wave < TILES_M) {
        Amat = insum; W = A; out = m_ws;
        row0 = (wave / NT) * 16; col0 = (wave % NT) * 16;
    } else if (wave < 2 * TILES_M) {
        int t = wave - TILES_M;
        Amat = insum; W = C; out = c_ws;
        row0 = (t / NT) * 16; col0 = (t % NT) * 16;
    } else {
        int t = wave - 2 * TILES_M;
        if (t >= TILES_U) return;
        Amat = qsum; W = B; out = u_ws;
        row0 = (t / NT) * 16; col0 = (t % NT) * 16;
    }

    int half = lane >> 4;          // 0: K pair (k,k+1); 1: (k+2,k+3)
    int l16  = lane & 15;

    v8f acc = {};
#pragma unroll 4
    for (int k = 0; k < EMB; k += 4) {
        const float* ap = Amat + (size_t)(row0 + l16) * EMB + k + half * 2;
        v2f a; a.x = ap[0]; a.y = ap[1];
        const float* bp = W + (size_t)(k + half * 2) * HID + col0 + l16;
        v2f b; b.x = bp[0]; b.y = bp[HID];
        // 8 args: (neg_a, A, neg_b, B, c_mod, C, reuse_a, reuse_b)
        acc = __builtin_amdgcn_wmma_f32_16x16x4_f32(
            false, a, false, b, (short)0, acc, false, false);
    }

#pragma unroll
    for (int vg = 0; vg < 8; ++vg) {
        out[(size_t)(row0 + vg + half * 8) * HID + col0 + l16] = acc[vg];
    }
}

// ---------------------------------------------------------------------------
// Kernel 3: 3 memory hops. One block per batch element, u in LDS.
//   logits[j] = u . m[j,b,:]; p = softmax(logits); u += sum_j p_j * c[j,b,:]
// m/c are L2-resident (3.2 MB total), traffic per block ~51 KB/hop.
// ---------------------------------------------------------------------------
__global__ void memn2n_hops(const float* __restrict__ m_ws,
                            const float* __restrict__ c_ws,
                            const float* __restrict__ u_ws,
                            float* __restrict__ out)
{
    __shared__ float u_s[HID];
    __shared__ float logit_s[MEM_SIZE];

    int b = blockIdx.x;
    int t = threadIdx.x;                 // 0..127

    u_s[t] = u_ws[b * HID + t];
    __syncthreads();

    for (int hop = 0; hop < HOPS; ++hop) {
        if (t < MEM_SIZE) {
            const float* mrow = m_ws + (size_t)(t * BATCH + b) * HID;
            float acc = 0.f;
#pragma unroll 8
            for (int h = 0; h < HID; ++h) acc += u_s[h] * mrow[h];
            logit_s[t] = acc;
        }
        __syncthreads();

        // redundant per-thread softmax over 50 entries (tiny)
        float mx = -1e30f;
        for (int j = 0; j < MEM_SIZE; ++j) mx = fmaxf(mx, logit_s[j]);
        float den = 0.f;
        for (int j = 0; j < MEM_SIZE; ++j) den += expf(logit_s[j] - mx);
        float inv = 1.f / den;

        float o = 0.f;
        for (int j = 0; j < MEM_SIZE; ++j) {
            float p = expf(logit_s[j] - mx) * inv;
            o += p * c_ws[(size_t)(j * BATCH + b) * HID + t];
        }
        __syncthreads();
        u_s[t] += o;
        __syncthreads();
    }

    out[b * HID + t] = u_s[t];
}

// ---------------------------------------------------------------------------
extern "C" void kernel_launch(void* const* d_in, const int* in_sizes, int n_in,
                              void* d_out, int out_size, void* d_ws, size_t ws_size,
                              hipStream_t stream)
{
    const float* inputs    = (const float*)d_in[0];
    const float* questions = (const float*)d_in[1];
    const float* A         = (const float*)d_in[2];
    const float* B         = (const float*)d_in[3];
    const float* C         = (const float*)d_in[4];
    float* out = (float*)d_out;

    float* ws    = (float*)d_ws;
    float* insum = ws;                                   // 3200*256
    float* qsum  = insum + MEM_SIZE * BATCH * EMB;       // 64*256
    float* m_ws  = qsum  + BATCH * EMB;                  // 3200*128
    float* c_ws  = m_ws  + MEM_SIZE * BATCH * HID;       // 3200*128
    float* u_ws  = c_ws  + MEM_SIZE * BATCH * HID;       // 64*128

    // 3264 rows total, 4 rows per block
    memn2n_reduce_len<<<816, 256, 0, stream>>>(inputs, questions, insum, qsum);

    // 1600 + 1600 + 32 = 3232 wave-tiles, 4 waves per 128-thread block
    memn2n_wmma_gemm<<<808, 128, 0, stream>>>(insum, qsum, A, B, C,
                                              m_ws, c_ws, u_ws);

    memn2n_hops<<<BATCH, HID, 0, stream>>>(m_ws, c_ws, u_ws, out);
}